// inhomNeuralSigKer_5789615915576
// MI455X (gfx1250) — compile-verified
//
#include <hip/hip_runtime.h>

// inhomogeneous neural signature kernel, id_phi defaults:
//   K[i,j,t] = prod_{t'<t} (1 + <dX_i,t', dY_j,t'> / dt),   dt = 1/1023
// kernel1: per-segment local cumprod, 16x16 (i,j) tile per wave, f32 WMMA over D=64,
//          async global->LDS tile pipeline (3-deep ring), t-coalesced output staging.
// kernel2: exclusive prefix product over segment totals (from d_ws) + in-place scale.

typedef float v2f __attribute__((ext_vector_type(2)));
typedef float v8f __attribute__((ext_vector_type(8)));
typedef int   v4i __attribute__((ext_vector_type(4)));

#define BATCH 128
#define TLEN  1024
#define NSTEP 1023      // number of increments
#define DDIM  64
#define TS    32        // steps per segment
#define NSEG  32        // ceil(1023/32); last segment has 31 steps
#define XPAD  68        // row stride (floats) for X/Y LDS tiles: 272B, 16B aligned
#define KPAD  33        // row stride (floats) for K staging (odd => conflict-free b32 writes)

#if defined(__has_builtin)
#  if __has_builtin(__builtin_amdgcn_global_load_async_to_lds_b128)
#    define HAVE_ASYNC_LDS 1
#  endif
#endif
#ifndef HAVE_ASYNC_LDS
#  define HAVE_ASYNC_LDS 0
#endif

#define AS_GLOBAL __attribute__((address_space(1)))
#define AS_LOCAL  __attribute__((address_space(3)))

// ---------------- Kernel 1: one wave per (16i x 16j tile, segment) ----------------
__global__ __launch_bounds__(32)
void sigker_seg_kernel(const float* __restrict__ X, const float* __restrict__ Y,
                       float* __restrict__ out, float* __restrict__ segtot) {
  __shared__ alignas(16) float Xs[3][16][XPAD];   // 3-deep ring over t
  __shared__ alignas(16) float Ys[3][16][XPAD];
  __shared__ alignas(16) float Kb[256 * KPAD];

  const int i0 = blockIdx.x * 16;
  const int j0 = blockIdx.y * 16;
  const int s  = blockIdx.z;
  const int t0 = s * TS;
  const int tEnd = (t0 + TS < NSTEP) ? (t0 + TS) : NSTEP;   // exclusive step bound
  const int lane = threadIdx.x;
  const int row  = lane & 15;     // A: M row / B: N col / tile row for loads
  const int half = lane >> 4;     // selects K-pair (A/B) and M-half (C/D)
  const float inv_dt = 1023.0f;

  const float* xrow = X + (size_t)(i0 + row) * TLEN * DDIM;
  const float* yrow = Y + (size_t)(j0 + row) * TLEN * DDIM;

  // Issue loads of the 16x64 X/Y tiles at time index t into ring buffer b.
  // Each lane moves 8 float4: lanes<16 cover floats 0..31 of its row, lanes>=16 cover 32..63.
  auto issue_tiles = [&](int t, int b) {
    const float4* xs = (const float4*)(xrow + (size_t)t * DDIM);
    const float4* ys = (const float4*)(yrow + (size_t)t * DDIM);
#pragma unroll
    for (int q = 0; q < 8; ++q) {
      const int c = half * 8 + q;
#if HAVE_ASYNC_LDS
      __builtin_amdgcn_global_load_async_to_lds_b128(
          (AS_GLOBAL v4i*)(xs + c), (AS_LOCAL v4i*)&Xs[b][row][c * 4], 0, 0);
      __builtin_amdgcn_global_load_async_to_lds_b128(
          (AS_GLOBAL v4i*)(ys + c), (AS_LOCAL v4i*)&Ys[b][row][c * 4], 0, 0);
#else
      *(float4*)&Xs[b][row][c * 4] = xs[c];
      *(float4*)&Ys[b][row][c * 4] = ys[c];
#endif
    }
  };

  // Wait until tiles needed this step are resident. When a prefetch batch
  // (16 async ops) was just issued, allow it to remain outstanding.
  auto wait_tiles = [&](bool prefetched) {
#if HAVE_ASYNC_LDS
    if (prefetched) asm volatile("s_wait_asynccnt 0x10" ::: "memory");
    else            asm volatile("s_wait_asynccnt 0x0"  ::: "memory");
#else
    (void)prefetched;
    asm volatile("s_wait_dscnt 0x0" ::: "memory");
#endif
  };

  issue_tiles(t0,     t0 % 3);
  issue_tiles(t0 + 1, (t0 + 1) % 3);

  v8f kreg;                       // running local cumulative product (starts at 1)
#pragma unroll
  for (int v = 0; v < 8; ++v) kreg[v] = 1.0f;

  for (int t = t0; t < tEnd; ++t) {
    const int cb = t % 3, nb = (t + 1) % 3;
    const bool pf = (t + 2 <= tEnd);
    if (pf) issue_tiles(t + 2, (t + 2) % 3);    // overlap with this step's compute
    wait_tiles(pf);

    // Two independent WMMA chains (even/odd K-chunks) to halve D->C dependency depth.
    v8f acc0 = {0.f, 0.f, 0.f, 0.f, 0.f, 0.f, 0.f, 0.f};
    v8f acc1 = {0.f, 0.f, 0.f, 0.f, 0.f, 0.f, 0.f, 0.f};
#pragma unroll
    for (int k = 0; k < 16; k += 2) {
      const int off0 = k * 4 + half * 2;
      const int off1 = off0 + 4;
      v2f a0, b0, a1, b1;
      a0.x = Xs[nb][row][off0]     - Xs[cb][row][off0];
      a0.y = Xs[nb][row][off0 + 1] - Xs[cb][row][off0 + 1];
      b0.x = Ys[nb][row][off0]     - Ys[cb][row][off0];
      b0.y = Ys[nb][row][off0 + 1] - Ys[cb][row][off0 + 1];
      a1.x = Xs[nb][row][off1]     - Xs[cb][row][off1];
      a1.y = Xs[nb][row][off1 + 1] - Xs[cb][row][off1 + 1];
      b1.x = Ys[nb][row][off1]     - Ys[cb][row][off1];
      b1.y = Ys[nb][row][off1 + 1] - Ys[cb][row][off1 + 1];
      acc0 = __builtin_amdgcn_wmma_f32_16x16x4_f32(
          false, a0, false, b0, (short)0, acc0, false, false);
      acc1 = __builtin_amdgcn_wmma_f32_16x16x4_f32(
          false, a1, false, b1, (short)0, acc1, false, false);
    }

    const int tl = t - t0;
#pragma unroll
    for (int v = 0; v < 8; ++v) {
      kreg[v] *= (1.0f + (acc0[v] + acc1[v]) * inv_dt);   // multiplicative scan step
      Kb[((v + half * 8) * 16 + row) * KPAD + tl] = kreg[v];
    }
  }

  // Segment total (unscaled) for the prefix pass
#pragma unroll
  for (int v = 0; v < 8; ++v) {
    const int M = v + half * 8;
    segtot[((size_t)(i0 + M) * BATCH + (size_t)(j0 + row)) * NSEG + s] = kreg[v];
  }

  // Write local cumprods with t as the coalesced (fastest) axis.
  const int nsteps = tEnd - t0;
  for (int p = 0; p < 256; ++p) {
    if (lane < nsteps) {
      const int M = p >> 4, N = p & 15;
      out[((size_t)(i0 + M) * BATCH + (size_t)(j0 + N)) * TLEN + (t0 + 1 + lane)] =
          Kb[p * KPAD + lane];
    }
  }
}

// ---------------- Kernel 2: prefix over segment totals, scale in place ----------------
__global__ __launch_bounds__(256)
void sigker_scale_kernel(float* __restrict__ out, const float* __restrict__ segtot) {
  const int i0 = blockIdx.x * 16;
  const int j0 = blockIdx.y * 16;
  const int s  = blockIdx.z;
  const int tid = threadIdx.x;
  const size_t pair = (size_t)(i0 + (tid >> 4)) * BATCH + (size_t)(j0 + (tid & 15));

  float P = 1.0f;                                // exclusive prefix product
  const float* st = segtot + pair * NSEG;
  for (int sb = 0; sb < s; ++sb) P *= st[sb];

  const int t0 = s * TS;
  const int nsteps = (t0 + TS < NSTEP) ? TS : (NSTEP - t0);
  float* o = out + pair * TLEN;
  if (s == 0) o[0] = 1.0f;                       // K[:, :, 0] = VAR_0
  for (int tl = 0; tl < nsteps; ++tl) o[t0 + 1 + tl] *= P;
}

// ---------------- Launch ----------------
extern "C" void kernel_launch(void* const* d_in, const int* in_sizes, int n_in,
                              void* d_out, int out_size, void* d_ws, size_t ws_size,
                              hipStream_t stream) {
  const float* X = (const float*)d_in[0];   // (128, 1024, 64) f32
  const float* Y = (const float*)d_in[1];   // (128, 1024, 64) f32
  float* out = (float*)d_out;               // (128, 128, 1024) f32
  float* segtot = (float*)d_ws;             // 128*128*NSEG floats = 2 MB scratch

  dim3 grid(BATCH / 16, BATCH / 16, NSEG);
  sigker_seg_kernel<<<grid, 32, 0, stream>>>(X, Y, out, segtot);
  sigker_scale_kernel<<<grid, 256, 0, stream>>>(out, segtot);
}